// ConvolutionalCapsuleLayer_87273735455245
// MI455X (gfx1250) — compile-verified
//
#include <hip/hip_runtime.h>
#include <hip/hip_bf16.h>

// ConvolutionalCapsuleLayer with dynamic routing — fully fused, one pass over HBM.
// u_hat (803 MB fp32) is never materialized: each workgroup owns a 16-row
// (w,b) tile, keeps xw + routing state in LDS, and recomputes u_hat fragments
// with V_WMMA_F32_16X16X4_F32 (f32 in / f32 acc — matches fp32 reference).

#define B_   4
#define L_   1024
#define FP_  32
#define DP_  8
#define KSZ  3
#define N_   (KSZ * FP_)      // 96
#define F_   32
#define E_   16
#define W_   (L_ - KSZ + 1)   // 1022
#define WB_  (W_ * B_)        // 4088
#define ROWS 16
#define NTHREADS 256
#define F_PER_WAVE 4          // 8 waves x 4 f-columns = 32 f
#define ROUT_ITERS 3

typedef float v2f __attribute__((ext_vector_type(2)));
typedef float v8f __attribute__((ext_vector_type(8)));

// LDS partition (floats). xw padded to stride 10: distinct banks for the 16
// m-lanes (m*10 mod 64 all distinct) and even offsets keep ds_load_b64 aligned.
#define XW_STRIDE 10
#define XW_FLOATS (N_ * ROWS * XW_STRIDE)   // 15360 (60 KB)
#define B_FLOATS  (ROWS * N_ * F_)          // 49152 (192 KB)
#define ST_FLOATS (ROWS * N_ * 2)           // 3072  (12 KB)
#define S_FLOATS  (ROWS * F_ * E_)          // 8192  (32 KB)
#define LDS_FLOATS (XW_FLOATS + B_FLOATS + ST_FLOATS + S_FLOATS)
#define LDS_BYTES  (LDS_FLOATS * 4)         // 303104 B < 320 KB/WGP

__global__ void __launch_bounds__(NTHREADS, 1)
capsule_routing_fused(const float* __restrict__ x,
                      const float* __restrict__ weight,
                      const float* __restrict__ bias,
                      float* __restrict__ out)
{
    extern __shared__ float smem[];
    float* lds_xw = smem;                    // [n][m][XW_STRIDE]
    float* lds_b  = smem + XW_FLOATS;        // [m][n][f]
    float* lds_st = lds_b + B_FLOATS;        // [m][n][{max, 1/sum}]
    float* lds_s  = lds_st + ST_FLOATS;      // [m][f][e]  (s, then v in place)

    const int tid  = threadIdx.x;
    const int lane = tid & 31;
    const int wave = tid >> 5;
    const int half = lane >> 4;              // 0: lanes 0-15, 1: lanes 16-31
    const int col  = lane & 15;              // e for B/C/D fragments, m for A
    const int ka   = half * 2;               // K pair base: {0,1} / {2,3}
    const int row0 = blockIdx.x * ROWS;
    const int f_base = wave * F_PER_WAVE;

    // ---- Phase 0: gather xw tile into LDS (coalesced over fp,d), zero b ----
    for (int idx = tid; idx < ROWS * N_ * DP_; idx += NTHREADS) {
        int m = idx / (N_ * DP_);
        int t = idx - m * (N_ * DP_);
        int n = t >> 3;
        int d = t & 7;
        int r = row0 + m; if (r >= WB_) r = WB_ - 1;   // clamp partial tile
        int b = r / W_;
        int w = r - b * W_;
        int k  = n >> 5;
        int fp = n & 31;
        lds_xw[(n * ROWS + m) * XW_STRIDE + d] =
            x[((b * L_ + (w + k)) * FP_ + fp) * DP_ + d];
    }
    for (int idx = tid; idx < B_FLOATS; idx += NTHREADS) lds_b[idx] = 0.0f;
    __syncthreads();

    for (int it = 0; it < ROUT_ITERS; ++it) {
        // ---- softmax stats per (m,n): max over f and 1/sum(exp) ----
        for (int idx = tid; idx < ROWS * N_; idx += NTHREADS) {
            const float* bp = &lds_b[idx * F_];
            float mx = bp[0];
            for (int f = 1; f < F_; ++f) mx = fmaxf(mx, bp[f]);
            float ssum = 0.0f;
            for (int f = 0; f < F_; ++f) ssum += __expf(bp[f] - mx);
            lds_st[idx * 2 + 0] = mx;
            lds_st[idx * 2 + 1] = 1.0f / ssum;
        }
        __syncthreads();

        // ---- s-pass: s[m,f,:] = sum_n softmax(b)[m,n,f] * u_hat[m,n,f,:] ----
        v8f s_acc[F_PER_WAVE];
        #pragma unroll
        for (int j = 0; j < F_PER_WAVE; ++j)
            #pragma unroll
            for (int v = 0; v < 8; ++v) s_acc[j][v] = 0.0f;

        for (int n = 0; n < N_; ++n) {
            // A fragment (16x8 tile of xw, split over two 16x16x4 WMMAs)
            const float* ap = &lds_xw[(n * ROWS + col) * XW_STRIDE + ka];
            v2f a0 = *(const v2f*)(ap);        // K = ka, ka+1
            v2f a1 = *(const v2f*)(ap + 4);    // K = ka+4, ka+5
            float mx[8], is[8];
            #pragma unroll
            for (int v = 0; v < 8; ++v) {
                int m = v + half * 8;
                mx[v] = lds_st[(m * N_ + n) * 2 + 0];
                is[v] = lds_st[(m * N_ + n) * 2 + 1];
            }
            #pragma unroll
            for (int j = 0; j < F_PER_WAVE; ++j) {
                int f = f_base + j;
                const float* wp = weight + ((n * F_ + f) * DP_ + ka) * E_ + col;
                v2f b0 = { wp[0],      wp[E_] };       // d = ka,   ka+1
                v2f b1 = { wp[4 * E_], wp[5 * E_] };   // d = ka+4, ka+5
                float bval = bias[(n * F_ + f) * E_ + col];
                v8f acc;
                #pragma unroll
                for (int v = 0; v < 8; ++v) acc[v] = bval;  // C = bias (e-col)
                acc = __builtin_amdgcn_wmma_f32_16x16x4_f32(
                          false, a0, false, b0, (short)0, acc, false, false);
                acc = __builtin_amdgcn_wmma_f32_16x16x4_f32(
                          false, a1, false, b1, (short)0, acc, false, false);
                #pragma unroll
                for (int v = 0; v < 8; ++v) {
                    int m = v + half * 8;
                    float bb = lds_b[(m * N_ + n) * F_ + f];
                    float c  = __expf(bb - mx[v]) * is[v];
                    s_acc[j][v] += c * acc[v];
                }
            }
        }
        #pragma unroll
        for (int j = 0; j < F_PER_WAVE; ++j) {
            int f = f_base + j;
            #pragma unroll
            for (int v = 0; v < 8; ++v)
                lds_s[((v + half * 8) * F_ + f) * E_ + col] = s_acc[j][v];
        }
        __syncthreads();

        // ---- squash in place: v = s * sm/((1+sm)*sqrt(sm)) ----
        for (int idx = tid; idx < ROWS * F_; idx += NTHREADS) {
            float* sp = &lds_s[idx * E_];
            float sm = 0.0f;
            for (int q = 0; q < E_; ++q) sm += sp[q] * sp[q];
            float scale = (sm / (1.0f + sm)) * rsqrtf(sm);
            for (int q = 0; q < E_; ++q) sp[q] *= scale;
        }
        __syncthreads();

        if (it < ROUT_ITERS - 1) {
            // ---- b-update: b[m,n,f] += sum_e v[m,f,e] * u_hat[m,n,f,e] ----
            for (int n = 0; n < N_; ++n) {
                const float* ap = &lds_xw[(n * ROWS + col) * XW_STRIDE + ka];
                v2f a0 = *(const v2f*)(ap);
                v2f a1 = *(const v2f*)(ap + 4);
                #pragma unroll
                for (int j = 0; j < F_PER_WAVE; ++j) {
                    int f = f_base + j;
                    const float* wp = weight + ((n * F_ + f) * DP_ + ka) * E_ + col;
                    v2f b0 = { wp[0],      wp[E_] };
                    v2f b1 = { wp[4 * E_], wp[5 * E_] };
                    float bval = bias[(n * F_ + f) * E_ + col];
                    v8f acc;
                    #pragma unroll
                    for (int v = 0; v < 8; ++v) acc[v] = bval;
                    acc = __builtin_amdgcn_wmma_f32_16x16x4_f32(
                              false, a0, false, b0, (short)0, acc, false, false);
                    acc = __builtin_amdgcn_wmma_f32_16x16x4_f32(
                              false, a1, false, b1, (short)0, acc, false, false);
                    #pragma unroll
                    for (int v = 0; v < 8; ++v) {
                        int m = v + half * 8;
                        // dot over e: 16-lane half-wave reduction (xor<16 stays in half)
                        float p = lds_s[(m * F_ + f) * E_ + col] * acc[v];
                        p += __shfl_xor(p, 1, 32);
                        p += __shfl_xor(p, 2, 32);
                        p += __shfl_xor(p, 4, 32);
                        p += __shfl_xor(p, 8, 32);
                        if (col == 0)
                            lds_b[(m * N_ + n) * F_ + f] += p;
                    }
                }
            }
            __syncthreads();
        }
    }

    // ---- write v -> out[b][w][f][e] == out[r*512 + f*16 + e], r = b*W + w ----
    for (int idx = tid; idx < ROWS * F_ * E_; idx += NTHREADS) {
        int m = idx / (F_ * E_);
        int r = row0 + m;
        if (r < WB_)
            out[r * (F_ * E_) + (idx - m * (F_ * E_))] = lds_s[idx];
    }
}

extern "C" void kernel_launch(void* const* d_in, const int* in_sizes, int n_in,
                              void* d_out, int out_size, void* d_ws, size_t ws_size,
                              hipStream_t stream) {
    (void)in_sizes; (void)n_in; (void)d_ws; (void)ws_size; (void)out_size;
    const float* x      = (const float*)d_in[0];
    const float* weight = (const float*)d_in[1];
    const float* bias   = (const float*)d_in[2];
    float* out = (float*)d_out;

    dim3 grid((WB_ + ROWS - 1) / ROWS);   // 256 workgroups, 16 rows each
    dim3 block(NTHREADS);                 // 8 wave32 waves
    capsule_routing_fused<<<grid, block, LDS_BYTES, stream>>>(x, weight, bias, out);
}